// GlobalAttentionPool_18021682774957
// MI455X (gfx1250) — compile-verified
//
#include <hip/hip_runtime.h>
#include <hip/hip_bf16.h>
#include <math.h>

typedef __attribute__((ext_vector_type(2))) float v2f;
typedef __attribute__((ext_vector_type(8))) float v8f;

#define HID 128

// ---------------------------------------------------------------------------
// Kernel 1: dual rank-1 projection via WMMA f32 16x16x4.
//   mrel[i] = x[i,:] . W_rel      (D column 0)
//   sarr[i] = x[i,:] . W_root + b (D column 1, bias folded in)
// One wave32 handles a 16-node tile; 32 accumulating WMMAs sweep K=128.
// A layout (ISA 7.12.2, 32-bit A 16x4): lane m in 0..15 -> row M=m,
//   VGPR0/VGPR1 = K {0,1} for lanes 0..15, K {2,3} for lanes 16..31.
// B layout (rows striped across lanes): lane n&15 supplies column n,
//   half h supplies rows K = {2h, 2h+1}.
// D layout: lane = column N (0..15 twice); c[j] = D[M = 8*half + j][N].
//
// B is built branch-free: every lane loads from a valid base pointer
// (Wrel or Wroot, selected once) and multiplies by a 0/1 mask for
// columns >= 2. This keeps the inner loop as pure load->wmma.
// ---------------------------------------------------------------------------
__global__ void k_dots(const float* __restrict__ x,
                       const float* __restrict__ Wrel,
                       const float* __restrict__ Wroot,
                       const float* __restrict__ brel,
                       float* __restrict__ mrel,
                       float* __restrict__ sarr,
                       int ntiles, int N)
{
    const int wave = threadIdx.x >> 5;
    const int lane = threadIdx.x & 31;
    const int tile = blockIdx.x * (blockDim.x >> 5) + wave;
    if (tile >= ntiles) return;                 // wave-uniform: EXEC stays full

    const int row  = lane & 15;
    const int half = lane >> 4;                 // 0 -> K pair {0,1}, 1 -> {2,3}
    const float bias = brel[0];

    int nodeIdx = tile * 16 + row;
    if (nodeIdx >= N) nodeIdx = N - 1;          // clamp loads on partial tile
    const float* __restrict__ xrow = x + (size_t)nodeIdx * HID;

    // Branch-free B operand setup (hoisted out of the K loop):
    const float* __restrict__ wbase = (row == 1) ? Wroot : Wrel;  // always valid
    const float msk = (row < 2) ? 1.0f : 0.0f;                    // zero cols >= 2

    v8f c = {};
    #pragma unroll
    for (int ko = 0; ko < HID / 4; ++ko) {
        const int kb = ko * 4 + half * 2;
        v2f a;
        a.x = xrow[kb];
        a.y = xrow[kb + 1];
        v2f b;
        b.x = msk * wbase[kb];
        b.y = msk * wbase[kb + 1];
        c = __builtin_amdgcn_wmma_f32_16x16x4_f32(
                /*neg_a=*/false, a, /*neg_b=*/false, b,
                /*c_mod=*/(short)0, c, /*reuse_a=*/false, /*reuse_b=*/false);
    }

    const int node0 = tile * 16 + (half ? 8 : 0);
    const bool fullTile = (tile * 16 + 16) <= N;   // wave-uniform
    if (fullTile) {
        if (row == 0) {                         // lanes 0,16: column 0 = rel dot
            #pragma unroll
            for (int j = 0; j < 8; ++j) mrel[node0 + j] = c[j];
        } else if (row == 1) {                  // lanes 1,17: column 1 = root dot
            #pragma unroll
            for (int j = 0; j < 8; ++j) sarr[node0 + j] = c[j] + bias;
        }
    } else {
        if (row == 0) {
            #pragma unroll
            for (int j = 0; j < 8; ++j)
                if (node0 + j < N) mrel[node0 + j] = c[j];
        } else if (row == 1) {
            #pragma unroll
            for (int j = 0; j < 8; ++j)
                if (node0 + j < N) sarr[node0 + j] = c[j] + bias;
        }
    }
}

// ---------------------------------------------------------------------------
// Kernel 2: edge scatter  sarr[dst] += mrel[src]
// ---------------------------------------------------------------------------
__global__ void k_scatter(const int* __restrict__ ei, int E,
                          const float* __restrict__ mrel,
                          float* __restrict__ sarr)
{
    const int e = blockIdx.x * blockDim.x + threadIdx.x;
    if (e < E) {
        const int s = ei[e];
        const int d = ei[E + e];
        atomicAdd(&sarr[d], mrel[s]);
    }
}

// ---------------------------------------------------------------------------
// Kernel 3: per-graph softmax + weighted pooling. One block per graph.
// batch is sorted -> binary search node range; all reductions in LDS,
// no global atomics; writes its 128-float output row directly.
// ---------------------------------------------------------------------------
__global__ void k_pool(const float* __restrict__ x,
                       const int* __restrict__ batch,
                       float* __restrict__ sarr,   // in: s ; scratch: z
                       float* __restrict__ out,
                       int N)
{
    __shared__ float red[256];
    const int g = blockIdx.x;
    const int t = threadIdx.x;

    // lower_bound(batch, g) and lower_bound(batch, g+1)
    int lo = 0, hi = N;
    while (lo < hi) { int mid = (lo + hi) >> 1; if (batch[mid] < g) lo = mid + 1; else hi = mid; }
    const int start = lo;
    hi = N;
    while (lo < hi) { int mid = (lo + hi) >> 1; if (batch[mid] < g + 1) lo = mid + 1; else hi = mid; }
    const int end = lo;

    // Phase A: segment max
    float mx = -INFINITY;
    for (int n = start + t; n < end; n += 256) mx = fmaxf(mx, sarr[n]);
    red[t] = mx; __syncthreads();
    for (int s = 128; s > 0; s >>= 1) {
        if (t < s) red[t] = fmaxf(red[t], red[t + s]);
        __syncthreads();
    }
    mx = red[0]; __syncthreads();

    // Phase B: z = exp(s - mx), denom = sum z   (cache z back into sarr:
    // [start,end) is owned exclusively by this block)
    float sum = 0.0f;
    for (int n = start + t; n < end; n += 256) {
        float z = expf(sarr[n] - mx);
        sarr[n] = z;
        sum += z;
    }
    red[t] = sum; __syncthreads();
    for (int s = 128; s > 0; s >>= 1) {
        if (t < s) red[t] += red[t + s];
        __syncthreads();
    }
    const float denom = red[0]; __syncthreads();

    // Phase C: gx[g,f] = (1/denom) * sum_n z[n] * x[n,f]
    // threads 0..127 cover features of even-slice nodes, 128..255 odd slice
    const int f = t & (HID - 1);
    const int slice = t >> 7;
    float acc = 0.0f;
    for (int n = start + slice; n < end; n += 2)
        acc += x[(size_t)n * HID + f] * sarr[n];
    red[t] = acc; __syncthreads();
    if (t < HID) {
        float v = red[t] + red[t + HID];
        out[(size_t)g * HID + f] = (end > start) ? (v / denom) : 0.0f;
    }
}

// ---------------------------------------------------------------------------
// Host launcher
// ---------------------------------------------------------------------------
extern "C" void kernel_launch(void* const* d_in, const int* in_sizes, int n_in,
                              void* d_out, int out_size, void* d_ws, size_t ws_size,
                              hipStream_t stream)
{
    (void)n_in; (void)ws_size;

    const float* x     = (const float*)d_in[0];
    const int*   ei    = (const int*)  d_in[1];   // (2, E) flattened
    const int*   batch = (const int*)  d_in[2];
    const float* Wrel  = (const float*)d_in[3];
    const float* brel  = (const float*)d_in[4];
    const float* Wroot = (const float*)d_in[5];
    // d_in[6] = num_graphs device scalar; G derived from out_size instead.

    const int N = in_sizes[0] / HID;
    const int E = in_sizes[1] / 2;
    const int G = out_size / HID;

    float* mrel = (float*)d_ws;       // N floats
    float* sarr = mrel + N;           // N floats

    const int ntiles = (N + 15) / 16;
    const int wavesPerBlock = 8;      // 256 threads = 8 wave32
    const int blocks1 = (ntiles + wavesPerBlock - 1) / wavesPerBlock;

    k_dots<<<blocks1, 256, 0, stream>>>(x, Wrel, Wroot, brel, mrel, sarr, ntiles, N);
    k_scatter<<<(E + 255) / 256, 256, 0, stream>>>(ei, E, mrel, sarr);
    k_pool<<<G, 256, 0, stream>>>(x, batch, sarr, (float*)d_out, N);
}